// CrossAdjacencyMatrix_29283087024787
// MI455X (gfx1250) — compile-verified
//
#include <hip/hip_runtime.h>

typedef __attribute__((ext_vector_type(2))) float v2f;
typedef __attribute__((ext_vector_type(8))) float v8f;

#define D_EMB 128

// ---------------- Kernel: row L2-normalize (with zero padding) ----------------
// One wave (32 lanes) per row; 4 floats/lane = 128 = D_EMB.
__global__ void k_normalize_rows(const float* __restrict__ in, float* __restrict__ out,
                                 int rows_in, int rows_out) {
  int row = blockIdx.x;
  if (row >= rows_out) return;
  int lane = threadIdx.x;  // 0..31
  float4 v = make_float4(0.f, 0.f, 0.f, 0.f);
  if (row < rows_in) {
    v = ((const float4*)(in + (size_t)row * D_EMB))[lane];
  }
  float ss = v.x * v.x + v.y * v.y + v.z * v.z + v.w * v.w;
#pragma unroll
  for (int m = 16; m >= 1; m >>= 1) ss += __shfl_xor(ss, m, 32);
  float s = rsqrtf(ss + 1e-8f);
  float4 o = make_float4(v.x * s, v.y * s, v.z * s, v.w * s);
  ((float4*)(out + (size_t)row * D_EMB))[lane] = o;
}

// ---------------- Kernel: cosine-sim GEMM via V_WMMA_F32_16X16X4_F32 ----------
// sim = A_norm (n16 x 128) * B_norm^T (128 x n16). One wave per 16x16 tile.
// A frag (16x4 f32): lane l holds row M=l%16; VGPR0=K=2*(l/16), VGPR1=K+1.
// B frag (4x16 f32): lane l holds col N=l%16 (= row of B_norm); same K split.
// C/D (16x16 f32, 8 VGPRs): VGPR r -> M = r + 8*(l/16), N = l%16.
__global__ void k_wmma_sim(const float* __restrict__ A, const float* __restrict__ B,
                           float* __restrict__ sim, int n16) {
  int tm = blockIdx.x, tn = blockIdx.y;
  int l = threadIdx.x;        // 0..31, EXEC all ones (no divergence)
  int ml = l & 15;
  int hi = l >> 4;
  const float* ap = A + (size_t)(tm * 16 + ml) * D_EMB + 2 * hi;
  const float* bp = B + (size_t)(tn * 16 + ml) * D_EMB + 2 * hi;
  v8f c = {0.f, 0.f, 0.f, 0.f, 0.f, 0.f, 0.f, 0.f};
#pragma unroll 4
  for (int k = 0; k < D_EMB; k += 4) {
    v2f a, b;
    a.x = ap[k];     a.y = ap[k + 1];
    b.x = bp[k];     b.y = bp[k + 1];
    c = __builtin_amdgcn_wmma_f32_16x16x4_f32(false, a, false, b, (short)0, c,
                                              false, false);
  }
  int col = tn * 16 + ml;
#pragma unroll
  for (int r = 0; r < 8; ++r) {
    int row = tm * 16 + r + 8 * hi;
    sim[(size_t)row * n16 + col] = c[r];
  }
}

// ---------------- Kernels: row / column max-pool over sim ----------------
__global__ void k_rowmax(const float* __restrict__ sim, float* __restrict__ out,
                         int n, int n16) {
  int i = blockIdx.x;  // row < R_SR
  float m = -3.402823466e38f;
  for (int j = threadIdx.x; j < n; j += blockDim.x)
    m = fmaxf(m, sim[(size_t)i * n16 + j]);
  __shared__ float red[256];
  red[threadIdx.x] = m;
  __syncthreads();
  for (int s = blockDim.x >> 1; s > 0; s >>= 1) {
    if (threadIdx.x < s) red[threadIdx.x] = fmaxf(red[threadIdx.x], red[threadIdx.x + s]);
    __syncthreads();
  }
  if (threadIdx.x == 0) out[i] = red[0];
}

__global__ void k_colmax(const float* __restrict__ sim, float* __restrict__ out,
                         int n, int n16) {
  int j = blockIdx.x;  // col < R_TG
  float m = -3.402823466e38f;
  for (int i = threadIdx.x; i < n; i += blockDim.x)
    m = fmaxf(m, sim[(size_t)i * n16 + j]);
  __shared__ float red[256];
  red[threadIdx.x] = m;
  __syncthreads();
  for (int s = blockDim.x >> 1; s > 0; s >>= 1) {
    if (threadIdx.x < s) red[threadIdx.x] = fmaxf(red[threadIdx.x], red[threadIdx.x + s]);
    __syncthreads();
  }
  if (threadIdx.x == 0) out[j] = red[0];
}

// ---------------- Kernel: init degrees to 1.0 (identity diagonal pre-added) --
__global__ void k_fill_ones(float* __restrict__ p, int n) {
  int i = blockIdx.x * blockDim.x + threadIdx.x;
  if (i < n) p[i] = 1.0f;
}

// ---------------- Kernel: edge pass 1 — vals + segment_sum(vals, head) -------
__global__ void k_edge_pass1(const float* __restrict__ conf, const float* __restrict__ imp,
                             const float* __restrict__ pca, const int* __restrict__ head,
                             float* __restrict__ v_out, float* __restrict__ deg, int E) {
  int i = (blockIdx.x * blockDim.x + threadIdx.x) * 4;
  if (i + 3 < E) {
    float4 c = *(const float4*)(conf + i);
    float4 m = *(const float4*)(imp + i);
    float4 p = *(const float4*)(pca + i);
    int4 h = *(const int4*)(head + i);
    float4 v = make_float4(c.x * m.x * p.x, c.y * m.y * p.y,
                           c.z * m.z * p.z, c.w * m.w * p.w);
    *(float4*)(v_out + i) = v;
    atomicAdd(deg + h.x, v.x);
    atomicAdd(deg + h.y, v.y);
    atomicAdd(deg + h.z, v.z);
    atomicAdd(deg + h.w, v.w);
  } else {
    for (; i < E; ++i) {
      float v = conf[i] * imp[i] * pca[i];
      v_out[i] = v;
      atomicAdd(deg + head[i], v);
    }
  }
}

// ---------------- Kernel: deg -> D^-1/2 in place ----------------
__global__ void k_deg_to_dis(float* __restrict__ deg, int n) {
  int i = blockIdx.x * blockDim.x + threadIdx.x;
  if (i < n) {
    float d = deg[i];
    deg[i] = (d > 0.f) ? rsqrtf(fmaxf(d, 1e-12f)) : 0.f;
  }
}

// ---------------- Kernel: edge pass 2 — v * dis[head] * dis[tail] ------------
__global__ void k_edge_pass2(const int* __restrict__ head, const int* __restrict__ tail,
                             const float* __restrict__ dis, float* __restrict__ adj, int E) {
  int i = (blockIdx.x * blockDim.x + threadIdx.x) * 4;
  if (i + 3 < E) {
    int4 h = *(const int4*)(head + i);
    int4 t = *(const int4*)(tail + i);
    float4 v = *(const float4*)(adj + i);
    v.x *= dis[h.x] * dis[t.x];
    v.y *= dis[h.y] * dis[t.y];
    v.z *= dis[h.z] * dis[t.z];
    v.w *= dis[h.w] * dis[t.w];
    *(float4*)(adj + i) = v;
  } else {
    for (; i < E; ++i) adj[i] *= dis[head[i]] * dis[tail[i]];
  }
}

// ---------------- Kernel: diagonal entries adj[E+i] = dis[i]^2 ---------------
__global__ void k_diag(const float* __restrict__ dis, float* __restrict__ adj_diag, int n) {
  int i = blockIdx.x * blockDim.x + threadIdx.x;
  if (i < n) {
    float d = dis[i];
    adj_diag[i] = d * d;
  }
}

extern "C" void kernel_launch(void* const* d_in, const int* in_sizes, int n_in,
                              void* d_out, int out_size, void* d_ws, size_t ws_size,
                              hipStream_t stream) {
  const float* rel_emb_sr = (const float*)d_in[0];
  const float* rel_emb_tg = (const float*)d_in[1];
  const float* conf_sr = (const float*)d_in[2];
  const float* imp_sr  = (const float*)d_in[3];
  const float* pca_sr  = (const float*)d_in[4];
  const float* conf_tg = (const float*)d_in[5];
  const float* imp_tg  = (const float*)d_in[6];
  const float* pca_tg  = (const float*)d_in[7];
  const int* head_sr = (const int*)d_in[8];
  const int* tail_sr = (const int*)d_in[9];
  // d_in[10] = relation_sr (unused by reference adjacency)
  const int* head_tg = (const int*)d_in[11];
  const int* tail_tg = (const int*)d_in[12];
  // d_in[13] = relation_tg (unused)

  const int R_SR = in_sizes[0] / D_EMB;   // 1000
  const int R_TG = in_sizes[1] / D_EMB;   // 1200
  const int E    = in_sizes[2];           // 4,000,000
  const int N    = (out_size - R_SR - R_TG) / 2 - E;  // 200,000
  const int n    = (R_SR > R_TG) ? R_SR : R_TG;       // 1200
  const int n16  = (n + 15) & ~15;
  const int nt   = n16 / 16;

  // Workspace layout (floats)
  float* ws = (float*)d_ws;
  float* a_norm = ws;                          // n16 * 128
  float* b_norm = a_norm + (size_t)n16 * D_EMB;
  float* sim    = b_norm + (size_t)n16 * D_EMB; // n16 * n16
  float* deg_sr = sim + (size_t)n16 * n16;      // N
  float* deg_tg = deg_sr + N;                   // N
  (void)ws_size; (void)n_in;

  // Output layout
  float* out    = (float*)d_out;
  float* adj_sr = out;                    // E + N
  float* adj_tg = out + (size_t)(E + N);  // E + N
  float* rw_sr  = out + (size_t)2 * (E + N);
  float* rw_tg  = rw_sr + R_SR;

  // ---- Relation weighting path (WMMA) ----
  k_normalize_rows<<<n16, 32, 0, stream>>>(rel_emb_sr, a_norm, R_SR, n16);
  k_normalize_rows<<<n16, 32, 0, stream>>>(rel_emb_tg, b_norm, R_TG, n16);
  k_wmma_sim<<<dim3(nt, nt), 32, 0, stream>>>(a_norm, b_norm, sim, n16);
  k_rowmax<<<R_SR, 256, 0, stream>>>(sim, rw_sr, n, n16);
  k_colmax<<<R_TG, 256, 0, stream>>>(sim, rw_tg, n, n16);

  // ---- Adjacency path (bandwidth-bound streaming) ----
  const int blk = 256;
  k_fill_ones<<<(2 * N + blk - 1) / blk, blk, 0, stream>>>(deg_sr, 2 * N);

  const int e4blocks = ((E + 3) / 4 + blk - 1) / blk;
  k_edge_pass1<<<e4blocks, blk, 0, stream>>>(conf_sr, imp_sr, pca_sr, head_sr,
                                             adj_sr, deg_sr, E);
  k_edge_pass1<<<e4blocks, blk, 0, stream>>>(conf_tg, imp_tg, pca_tg, head_tg,
                                             adj_tg, deg_tg, E);

  k_deg_to_dis<<<(2 * N + blk - 1) / blk, blk, 0, stream>>>(deg_sr, 2 * N);

  k_edge_pass2<<<e4blocks, blk, 0, stream>>>(head_sr, tail_sr, deg_sr, adj_sr, E);
  k_edge_pass2<<<e4blocks, blk, 0, stream>>>(head_tg, tail_tg, deg_tg, adj_tg, E);

  k_diag<<<(N + blk - 1) / blk, blk, 0, stream>>>(deg_sr, adj_sr + E, N);
  k_diag<<<(N + blk - 1) / blk, blk, 0, stream>>>(deg_tg, adj_tg + E, N);
}